// NodeTypeEmbedding_25623774888162
// MI455X (gfx1250) — compile-verified
//
#include <hip/hip_runtime.h>
#include <hip/hip_bf16.h>

typedef __attribute__((ext_vector_type(2))) float v2f;
typedef __attribute__((ext_vector_type(8))) float v8f;

#define NUM_TYPES 8
#define EMBED_DIM 32

// c0[v][t] = one_hot(node_type[v])[t]; c1 = 0
__global__ void k_init(const int* __restrict__ node_type,
                       float* __restrict__ c0, float* __restrict__ c1, int n_nodes) {
    int idx = blockIdx.x * blockDim.x + threadIdx.x;
    if (idx >= n_nodes * NUM_TYPES) return;
    int v = idx >> 3;
    int t = idx & 7;
    c0[idx] = (node_type[v] == t) ? 1.0f : 0.0f;
    c1[idx] = 0.0f;
}

__global__ void k_zero(float* __restrict__ p, int n) {
    int idx = blockIdx.x * blockDim.x + threadIdx.x;
    if (idx < n) p[idx] = 0.0f;
}

// One propagation round over both edge directions in 8-dim type space.
// 8 lanes per edge: lane handles one type-dim. Gather 32B coalesced per edge,
// two relaxed agent-scope f32 atomics per lane (u->v and v->u).
__global__ void k_edges(const int* __restrict__ src, const int* __restrict__ dst,
                        const float* __restrict__ cin, float* __restrict__ cout,
                        int n_edges) {
    long long gid = (long long)blockIdx.x * blockDim.x + threadIdx.x;
    int msg = (int)(gid >> 3);
    int lc  = (int)(gid & 7);
    if (msg >= n_edges) return;
    int u = src[msg];
    int v = dst[msg];
    float cu = cin[(size_t)u * NUM_TYPES + lc];
    float cv = cin[(size_t)v * NUM_TYPES + lc];
    __hip_atomic_fetch_add(&cout[(size_t)v * NUM_TYPES + lc], cu,
                           __ATOMIC_RELAXED, __HIP_MEMORY_SCOPE_AGENT);
    __hip_atomic_fetch_add(&cout[(size_t)u * NUM_TYPES + lc], cv,
                           __ATOMIC_RELAXED, __HIP_MEMORY_SCOPE_AGENT);
}

// out[0:n_rows, 0:32] = counts[0:n_rows, 0:8] @ W[8, 32]
// One wave32 computes one 16x16 output tile via V_WMMA_F32_16X16X4_F32,
// two K-steps (K=8). Exact f32 arithmetic.
__global__ void k_embed_gemm(const float* __restrict__ counts,
                             const float* __restrict__ W,
                             float* __restrict__ out, int n_rows) {
    const int wave = (blockIdx.x * blockDim.x + threadIdx.x) >> 5;
    const int lane = threadIdx.x & 31;
    const int n_mtiles = (n_rows + 15) >> 4;
    const int mt = wave >> 1;   // 16-row tile
    const int nt = wave & 1;    // 16-col tile (EMBED_DIM=32 -> 2 tiles)
    if (mt >= n_mtiles) return; // wave-uniform exit: EXEC all-ones at WMMA

    const int m     = lane & 15;        // row within tile (A), col within tile (B/D)
    const int khalf = (lane >> 4) & 1;  // lanes 16-31 hold upper K pair

    v8f acc = {0.f, 0.f, 0.f, 0.f, 0.f, 0.f, 0.f, 0.f};
    const float* crow = counts + (size_t)(mt * 16 + m) * NUM_TYPES;

#pragma unroll
    for (int s = 0; s < 2; ++s) {
        const int kb = s * 4 + khalf * 2; // global k of this lane's first element
        v2f a, b;
        // A 16x4 f32: lane = (k>>1)*16 + m, vgpr = k&1
        a.x = crow[kb];
        a.y = crow[kb + 1];
        // B 4x16 f32: lane = (k>>1)*16 + n, vgpr = k&1
        b.x = W[(size_t)kb       * EMBED_DIM + nt * 16 + m];
        b.y = W[(size_t)(kb + 1) * EMBED_DIM + nt * 16 + m];
        acc = __builtin_amdgcn_wmma_f32_16x16x4_f32(
            /*neg_a=*/false, a, /*neg_b=*/false, b,
            /*c_mod=*/(short)0, acc, /*reuse_a=*/false, /*reuse_b=*/false);
    }

    // D layout: acc[r] = row (r + 8*khalf) of the tile, col = lane&15.
    // Wave-uniform fast path for full tiles: 8 unconditional coalesced stores
    // (compiler folds into immediate-offset stores, no EXEC churn).
    if (mt * 16 + 16 <= n_rows) {
        float* obase = out + (size_t)(mt * 16 + khalf * 8) * EMBED_DIM + nt * 16 + m;
#pragma unroll
        for (int r = 0; r < 8; ++r) {
            obase[(size_t)r * EMBED_DIM] = acc[r];
        }
    } else { // only the final partial tile takes the predicated path
#pragma unroll
        for (int r = 0; r < 8; ++r) {
            int row = mt * 16 + khalf * 8 + r;
            if (row < n_rows) out[(size_t)row * EMBED_DIM + nt * 16 + m] = acc[r];
        }
    }
}

extern "C" void kernel_launch(void* const* d_in, const int* in_sizes, int n_in,
                              void* d_out, int out_size, void* d_ws, size_t ws_size,
                              hipStream_t stream) {
    const float* embed_table = (const float*)d_in[0]; // [8, 32]
    const int*   node_type   = (const int*)d_in[1];   // [N]
    const int*   src         = (const int*)d_in[2];   // [E]
    const int*   dst         = (const int*)d_in[3];   // [E]
    float*       out         = (float*)d_out;         // [N/8, 32]

    const int n_nodes = in_sizes[1];
    const int n_edges = in_sizes[2];
    const int n_rows  = out_size / EMBED_DIM;

    float* c0 = (float*)d_ws;                         // [N, 8]
    float* c1 = c0 + (size_t)n_nodes * NUM_TYPES;     // [N, 8]

    const int nc = n_nodes * NUM_TYPES;

    // init: c0 = one_hot(type), c1 = 0
    k_init<<<(nc + 255) / 256, 256, 0, stream>>>(node_type, c0, c1, n_nodes);

    // round 1: c1 += (A + A^T) c0
    {
        long long threads = (long long)n_edges * 8;
        int blocks = (int)((threads + 255) / 256);
        k_edges<<<blocks, 256, 0, stream>>>(src, dst, c0, c1, n_edges);
    }

    // round 2: c0 = 0; c0 += (A + A^T) c1
    k_zero<<<(nc + 255) / 256, 256, 0, stream>>>(c0, nc);
    {
        long long threads = (long long)n_edges * 8;
        int blocks = (int)((threads + 255) / 256);
        k_edges<<<blocks, 256, 0, stream>>>(src, dst, c1, c0, n_edges);
    }

    // out = c0[0:n_rows] @ W  via f32 WMMA
    {
        int n_mtiles = (n_rows + 15) / 16;
        int tiles = n_mtiles * 2;          // 2 N-tiles of 16 cols
        int waves_per_block = 256 / 32;    // 8
        int blocks = (tiles + waves_per_block - 1) / waves_per_block;
        k_embed_gemm<<<blocks, 256, 0, stream>>>(c0, embed_table, out, n_rows);
    }
}